// CDLossEval_31980326486603
// MI455X (gfx1250) — compile-verified
//
#include <hip/hip_runtime.h>
#include <hip/hip_bf16.h>
#include <math.h>

typedef __attribute__((ext_vector_type(2))) float v2f;
typedef __attribute__((ext_vector_type(8))) float v8f;

#define BATCH   16
#define NPTS    4096
#define NTILES  (NPTS / 16)      // 256 column tiles
#define WAVES_PER_BLOCK 8
#define ROWTILES_PER_BLOCK WAVES_PER_BLOCK

// For each batch b and each "row" point p in P:
//   dist[b,p] = min_q max(|p-q|^2, 0) = max( aa_p + min_q (bb_q - 2 p.q), 0 )
// (row-constant aa commutes with the column min; clamp commutes with min).
// V_WMMA_F32_16X16X4_F32 with K-slots:
//   A row = [ax, ay, az, 1],  B col = [-2bx, -2by, -2bz, bb]
//   -> product = bb - 2 a.b   (no per-tile epilogue adds at all)
// B columns are staged in LDS pre-scaled as (-2x, -2y, -2z, bb) so per-tile
// setup is a single dual cndmask.
__global__ __launch_bounds__(256) void cd_rowmin_kernel(
    const float* __restrict__ P,   // [B, NPTS, 3] row points
    const float* __restrict__ Q,   // [B, NPTS, 3] column points
    float* __restrict__ dist)      // [B, NPTS]
{
    __shared__ float4 qs[NPTS];    // 64 KB: (-2x, -2y, -2z, |q|^2) per column point

    const int b   = blockIdx.y;
    const int tid = threadIdx.x;

    const float* Qb = Q + (size_t)b * NPTS * 3;
    for (int p = tid; p < NPTS; p += 256) {
        const float x = Qb[p * 3 + 0];
        const float y = Qb[p * 3 + 1];
        const float z = Qb[p * 3 + 2];
        float4 v;
        v.x = -2.0f * x;
        v.y = -2.0f * y;
        v.z = -2.0f * z;
        v.w = x * x + y * y + z * z;
        qs[p] = v;
    }
    __syncthreads();

    const int wave = tid >> 5;
    const int lane = tid & 31;
    const int hi   = lane >> 4;        // which 16-lane half
    const int l16  = lane & 15;

    const int rowTile = blockIdx.x * ROWTILES_PER_BLOCK + wave;   // 0..255
    const int rowBase = rowTile * 16;

    // Load this wave's 16 A points; every lane reads point rowBase + (lane%16)
    const float* Pb = P + (size_t)b * NPTS * 3;
    const float ax = Pb[(rowBase + l16) * 3 + 0];
    const float ay = Pb[(rowBase + l16) * 3 + 1];
    const float az = Pb[(rowBase + l16) * 3 + 2];
    const float aa = ax * ax + ay * ay + az * az;   // added in the epilogue, not the loop

    // A-matrix 16x4 f32 layout: lanes 0-15 -> VGPR0=K0, VGPR1=K1 ; lanes 16-31 -> K2, K3
    v2f A;
    A.x = hi ? az : ax;
    A.y = hi ? 1.0f : ay;

    v8f dmin;
#pragma unroll
    for (int v = 0; v < 8; ++v) dmin[v] = 3.0e38f;

    for (int t = 0; t < NTILES; ++t) {
        const float4 qv = qs[t * 16 + l16];   // one ds_load_b128 per lane

        // B-matrix 4x16 f32 layout: lanes 0-15 -> VGPR0=K0, VGPR1=K1 ; lanes 16-31 -> K2, K3
        // Pre-scaled in LDS, so setup is just a half-select.
        v2f Bm;
        Bm.x = hi ? qv.z : qv.x;
        Bm.y = hi ? qv.w : qv.y;

        v8f acc = {};   // C = 0
        acc = __builtin_amdgcn_wmma_f32_16x16x4_f32(
            /*neg_a=*/false, A, /*neg_b=*/false, Bm,
            /*c_mod=*/(short)0, acc, /*reuse_a=*/false, /*reuse_b=*/false);

        // acc[v] (lane) = bb_col - 2 a.b for row = v + 8*hi, col = l16 of this tile.
#pragma unroll
        for (int v = 0; v < 8; ++v) {
            dmin[v] = fminf(dmin[v], acc[v]);
        }
    }

    // Row-min across the 16 lanes of each half (xor masks stay inside the half),
    // then add the deferred row norm aa and clamp to zero.
#pragma unroll
    for (int v = 0; v < 8; ++v) {
        float m = dmin[v];
        m = fminf(m, __shfl_xor(m, 1, 32));
        m = fminf(m, __shfl_xor(m, 2, 32));
        m = fminf(m, __shfl_xor(m, 4, 32));
        m = fminf(m, __shfl_xor(m, 8, 32));
        // Row r = v + 8*hi; its aa was loaded by lane l16 == r (low half holds all 16).
        const float aaRow = __shfl(aa, v + 8 * hi, 32);
        dmin[v] = fmaxf(m + aaRow, 0.0f);
    }

    if (l16 == 0) {
        float* drow = dist + (size_t)b * NPTS + rowBase + 8 * hi;
#pragma unroll
        for (int v = 0; v < 8; ++v) drow[v] = dmin[v];
    }
}

// Deterministic single-block reduction: out[0] = mean(dist1) + mean(dist2)
__global__ __launch_bounds__(256) void cd_finalize_kernel(
    const float* __restrict__ dist1,
    const float* __restrict__ dist2,
    float* __restrict__ out)
{
    __shared__ float red[256];
    const int tid = threadIdx.x;
    const int total = BATCH * NPTS;   // 65536

    float s1 = 0.0f, s2 = 0.0f;
    for (int i = tid; i < total; i += 256) s1 += dist1[i];
    for (int i = tid; i < total; i += 256) s2 += dist2[i];
    red[tid] = s1 * (1.0f / (float)total) + s2 * (1.0f / (float)total);
    __syncthreads();

    for (int off = 128; off > 0; off >>= 1) {
        if (tid < off) red[tid] += red[tid + off];
        __syncthreads();
    }
    if (tid == 0) out[0] = red[0];
}

extern "C" void kernel_launch(void* const* d_in, const int* in_sizes, int n_in,
                              void* d_out, int out_size, void* d_ws, size_t ws_size,
                              hipStream_t stream) {
    const float* pcs1 = (const float*)d_in[0];   // [16,4096,3]
    const float* pcs2 = (const float*)d_in[1];   // [16,4096,3]
    float* out   = (float*)d_out;                // [1 + 65536 + 65536]
    float* dist1 = out + 1;
    float* dist2 = out + 1 + BATCH * NPTS;

    dim3 block(256);
    dim3 grid(NTILES / ROWTILES_PER_BLOCK, BATCH);   // (32, 16)

    // dist1: rows = pcs1 points, cols = pcs2 (min over pcs2)
    cd_rowmin_kernel<<<grid, block, 0, stream>>>(pcs1, pcs2, dist1);
    // dist2: symmetric — swap roles (min over pcs1)
    cd_rowmin_kernel<<<grid, block, 0, stream>>>(pcs2, pcs1, dist2);
    // scalar mean
    cd_finalize_kernel<<<1, 256, 0, stream>>>(dist1, dist2, out);
}